// MemoryAccess_47674136985733
// MI455X (gfx1250) — compile-verified
//
#include <hip/hip_runtime.h>
#include <math.h>

#define Bx 64
#define Dx 512
#define Mx 1024
#define Wx 64
#define Rx 4
#define PD 480          // 471 projection outputs padded to 480 (30 x 16)
#define EPSf 1e-6f
#define LSTR 68         // LDS tile row stride: 68*4 bytes = 16B-aligned rows

typedef float v2f __attribute__((ext_vector_type(2)));
typedef float v8f __attribute__((ext_vector_type(8)));
typedef int   v4i_ __attribute__((ext_vector_type(4)));
#define AS1 __attribute__((address_space(1)))
#define AS3 __attribute__((address_space(3)))

__device__ __forceinline__ float sigmoidf_(float x) { return 1.0f / (1.0f + expf(-x)); }
__device__ __forceinline__ float softplusf_(float x) {
  return (x > 0.0f) ? (x + log1pf(expf(-x))) : log1pf(expf(x));
}

// ---------------------------------------------------------------------------
// 1) Pack the 10 weight matrices (D x o_i) into one 512x480 matrix + bias(480)
// Column layout: wv[0,64) ev[64,128) fg[128,132) ag[132] wg[133] rm[134,146)
//                ws[146] rs[147,151) wk[151,215) rk[215,471) pad[471,480)
// ---------------------------------------------------------------------------
__global__ void k_pack(const float* wv_w, const float* wv_b,
                       const float* ev_w, const float* ev_b,
                       const float* fg_w, const float* fg_b,
                       const float* ag_w, const float* ag_b,
                       const float* wg_w, const float* wg_b,
                       const float* rm_w, const float* rm_b,
                       const float* ws_w, const float* ws_b,
                       const float* rs_w, const float* rs_b,
                       const float* wk_w, const float* wk_b,
                       const float* rk_w, const float* rk_b,
                       float* __restrict__ Wc, float* __restrict__ bc) {
  int gid = blockIdx.x * blockDim.x + threadIdx.x;
  int NW_ELEMS = Dx * PD;
  if (gid < NW_ELEMS) {
    int k = gid / PD, c = gid % PD;
    const float* wp = nullptr; int width = 0, off = 0;
    if (c < 64)        { wp = wv_w; width = 64;  off = c; }
    else if (c < 128)  { wp = ev_w; width = 64;  off = c - 64; }
    else if (c < 132)  { wp = fg_w; width = 4;   off = c - 128; }
    else if (c < 133)  { wp = ag_w; width = 1;   off = 0; }
    else if (c < 134)  { wp = wg_w; width = 1;   off = 0; }
    else if (c < 146)  { wp = rm_w; width = 12;  off = c - 134; }
    else if (c < 147)  { wp = ws_w; width = 1;   off = 0; }
    else if (c < 151)  { wp = rs_w; width = 4;   off = c - 147; }
    else if (c < 215)  { wp = wk_w; width = 64;  off = c - 151; }
    else if (c < 471)  { wp = rk_w; width = 256; off = c - 215; }
    Wc[gid] = wp ? wp[k * width + off] : 0.0f;
  } else if (gid < NW_ELEMS + PD) {
    int c = gid - NW_ELEMS;
    const float* bp = nullptr; int off = 0;
    if (c < 64)        { bp = wv_b; off = c; }
    else if (c < 128)  { bp = ev_b; off = c - 64; }
    else if (c < 132)  { bp = fg_b; off = c - 128; }
    else if (c < 133)  { bp = ag_b; off = 0; }
    else if (c < 134)  { bp = wg_b; off = 0; }
    else if (c < 146)  { bp = rm_b; off = c - 134; }
    else if (c < 147)  { bp = ws_b; off = 0; }
    else if (c < 151)  { bp = rs_b; off = c - 147; }
    else if (c < 215)  { bp = wk_b; off = c - 151; }
    else if (c < 471)  { bp = rk_b; off = c - 215; }
    bc[c] = bp ? bp[off] : 0.0f;
  }
}

// ---------------------------------------------------------------------------
// 2) Projection GEMM: proj(64x480) = X(64x512) @ Wc(512x480) + bias
//    One wave32 per 16x16 output tile, V_WMMA_F32_16X16X4_F32 over K.
// ---------------------------------------------------------------------------
__global__ void k_gemm(const float* __restrict__ X, const float* __restrict__ Wc,
                       const float* __restrict__ bc, float* __restrict__ proj) {
  const int NT = PD / 16; // 30
  int tile = blockIdx.x;
  int mt = tile / NT, nt = tile % NT;
  int lane = threadIdx.x;
  int half = lane >> 4, l = lane & 15;
  int m0 = mt * 16, n0 = nt * 16;
  v8f acc = {0.f, 0.f, 0.f, 0.f, 0.f, 0.f, 0.f, 0.f};
  for (int k = 0; k < Dx; k += 4) {
    int ka = k + half * 2;
    v2f a, b;
    a[0] = X[(m0 + l) * Dx + ka];
    a[1] = X[(m0 + l) * Dx + ka + 1];
    b[0] = Wc[ka * PD + n0 + l];
    b[1] = Wc[(ka + 1) * PD + n0 + l];
    acc = __builtin_amdgcn_wmma_f32_16x16x4_f32(false, a, false, b, (short)0, acc,
                                                false, false);
  }
  for (int v = 0; v < 8; ++v) {
    int mm = m0 + v + half * 8;
    int nn = n0 + l;
    proj[mm * PD + nn] = acc[v] + bc[nn];
  }
}

// ---------------------------------------------------------------------------
// 3) In-place activations on proj
// ---------------------------------------------------------------------------
__global__ void k_act(float* __restrict__ proj) {
  int b = blockIdx.x, t = threadIdx.x;
  float* pr = proj + b * PD;
  if (t < 64) {
    pr[64 + t] = sigmoidf_(pr[64 + t]);                 // erase vectors
  } else if (t < 68) {
    pr[128 + (t - 64)] = sigmoidf_(pr[128 + (t - 64)]); // free gates
  } else if (t == 68) {
    pr[132] = sigmoidf_(pr[132]);                       // allocation gate
  } else if (t == 69) {
    pr[133] = sigmoidf_(pr[133]);                       // write gate
  } else if (t < 74) {
    int r = t - 70;                                     // read-mode softmax(3)
    float a0 = pr[134 + 3 * r], a1 = pr[135 + 3 * r], a2 = pr[136 + 3 * r];
    float mx = fmaxf(a0, fmaxf(a1, a2));
    float e0 = expf(a0 - mx), e1 = expf(a1 - mx), e2 = expf(a2 - mx);
    float inv = 1.0f / (e0 + e1 + e2);
    pr[134 + 3 * r] = e0 * inv; pr[135 + 3 * r] = e1 * inv; pr[136 + 3 * r] = e2 * inv;
  } else if (t == 74) {
    pr[146] = softplusf_(pr[146]);                      // write strength
  } else if (t < 79) {
    pr[147 + (t - 75)] = softplusf_(pr[147 + (t - 75)]); // read strengths
  }
}

// ---------------------------------------------------------------------------
// 4) usage = (prev_usage + (1-prev_usage)*prev_ww) * prod_r(1 - fg_r*prev_rw)
// ---------------------------------------------------------------------------
__global__ void k_usage(const float* __restrict__ pu, const float* __restrict__ pw,
                        const float* __restrict__ prw, const float* __restrict__ proj,
                        float* __restrict__ usage_out) {
  int b = blockIdx.x, t = threadIdx.x;
  const float* pr = proj + b * PD;
  float f0 = pr[128], f1 = pr[129], f2 = pr[130], f3 = pr[131];
  for (int m = t; m < Mx; m += 256) {
    float w = pw[b * Mx + m];
    float u = pu[b * Mx + m];
    float uaw = u + (1.0f - u) * w;
    float phi = (1.0f - f0 * prw[(b * Rx + 0) * Mx + m])
              * (1.0f - f1 * prw[(b * Rx + 1) * Mx + m])
              * (1.0f - f2 * prw[(b * Rx + 2) * Mx + m])
              * (1.0f - f3 * prw[(b * Rx + 3) * Mx + m]);
    usage_out[b * Mx + m] = uaw * phi;
  }
}

// ---------------------------------------------------------------------------
// Block reductions (blockDim == 256)
// ---------------------------------------------------------------------------
__device__ float blk_max256(float* red, float v) {
  int t = threadIdx.x;
  red[t] = v; __syncthreads();
  for (int s = 128; s > 0; s >>= 1) {
    if (t < s) red[t] = fmaxf(red[t], red[t + s]);
    __syncthreads();
  }
  float r = red[0]; __syncthreads();
  return r;
}
__device__ float blk_sum256(float* red, float v) {
  int t = threadIdx.x;
  red[t] = v; __syncthreads();
  for (int s = 128; s > 0; s >>= 1) {
    if (t < s) red[t] += red[t + s];
    __syncthreads();
  }
  float r = red[0]; __syncthreads();
  return r;
}

// ---------------------------------------------------------------------------
// 5/11) Cosine content weights: softmax_m( beta * <key,mem_m>/(|key||mem_m|) )
// grid = B*heads, block = 256
// ---------------------------------------------------------------------------
__global__ void k_cosine(const float* __restrict__ memsrc, const float* __restrict__ proj,
                         float* __restrict__ out, int heads, int key_off, int beta_off) {
  __shared__ float key[Wx];
  __shared__ float sim[Mx];
  __shared__ float red[256];
  int blk = blockIdx.x;
  int b = blk / heads, h = blk % heads;
  int t = threadIdx.x;
  const float* pr = proj + b * PD;
  if (t < Wx) key[t] = pr[key_off + h * Wx + t];
  __syncthreads();
  float beta = pr[beta_off + h];
  float kn2 = EPSf;
  for (int d = 0; d < Wx; ++d) kn2 += key[d] * key[d];
  float kn = sqrtf(kn2);
  for (int m = t; m < Mx; m += 256) {
    const float4* row4 = (const float4*)(memsrc + ((size_t)b * Mx + m) * Wx);
    float dot = 0.0f, nn = EPSf;
#pragma unroll
    for (int d4 = 0; d4 < Wx / 4; ++d4) {
      float4 mv = row4[d4];
      dot += key[d4 * 4 + 0] * mv.x + key[d4 * 4 + 1] * mv.y
           + key[d4 * 4 + 2] * mv.z + key[d4 * 4 + 3] * mv.w;
      nn += mv.x * mv.x + mv.y * mv.y + mv.z * mv.z + mv.w * mv.w;
    }
    sim[m] = dot / (kn * sqrtf(nn)) * beta;
  }
  __syncthreads();
  float lm = -3.4e38f;
  for (int m = t; m < Mx; m += 256) lm = fmaxf(lm, sim[m]);
  float mx = blk_max256(red, lm);
  float ls = 0.0f;
  for (int m = t; m < Mx; m += 256) { float e = expf(sim[m] - mx); sim[m] = e; ls += e; }
  __syncthreads();
  float inv = 1.0f / blk_sum256(red, ls);
  for (int m = t; m < Mx; m += 256)
    out[((size_t)b * heads + h) * Mx + m] = sim[m] * inv;
}

// ---------------------------------------------------------------------------
// 6) Allocation weights: bitonic argsort (stable via index tiebreak) of usage,
//    exclusive product scan, scatter back. One 1024-thread block per batch.
// ---------------------------------------------------------------------------
__global__ void k_alloc(const float* __restrict__ usage, float* __restrict__ waw) {
  __shared__ float val[Mx];
  __shared__ int   idx[Mx];
  __shared__ float scan[Mx];
  int b = blockIdx.x, t = threadIdx.x;
  val[t] = EPSf + (1.0f - EPSf) * usage[b * Mx + t];
  idx[t] = t;
  __syncthreads();
  for (int k = 2; k <= Mx; k <<= 1) {
    for (int j = k >> 1; j > 0; j >>= 1) {
      int ixj = t ^ j;
      if (ixj > t) {
        bool up = ((t & k) == 0);
        float v0 = val[t], v1 = val[ixj];
        int i0 = idx[t], i1 = idx[ixj];
        bool sw = up ? (v0 > v1 || (v0 == v1 && i0 > i1))
                     : (v0 < v1 || (v0 == v1 && i0 < i1));
        if (sw) { val[t] = v1; val[ixj] = v0; idx[t] = i1; idx[ixj] = i0; }
      }
      __syncthreads();
    }
  }
  scan[t] = val[t];
  __syncthreads();
  for (int off = 1; off < Mx; off <<= 1) {
    float x = (t >= off) ? scan[t - off] : 1.0f;
    __syncthreads();
    scan[t] = scan[t] * x;
    __syncthreads();
  }
  float ecp = (t == 0) ? 1.0f : scan[t - 1];
  waw[b * Mx + idx[t]] = (1.0f - val[t]) * ecp;
}

// ---------------------------------------------------------------------------
// 7) write_weights = wg*(ag*waw + (1-ag)*wcw); precedence = (1-sum(ww))*pp + ww
// ---------------------------------------------------------------------------
__global__ void k_ww_prec(const float* __restrict__ proj, const float* __restrict__ waw,
                          const float* __restrict__ wcw, const float* __restrict__ pp,
                          float* __restrict__ ww_out, float* __restrict__ prec_out) {
  __shared__ float red[256];
  int b = blockIdx.x, t = threadIdx.x;
  const float* pr = proj + b * PD;
  float ag = pr[132], wg = pr[133];
  float wloc[4]; float part = 0.0f;
  for (int i = 0; i < 4; ++i) {
    int m = t + i * 256;
    float w = wg * (ag * waw[b * Mx + m] + (1.0f - ag) * wcw[b * Mx + m]);
    wloc[i] = w; ww_out[b * Mx + m] = w; part += w;
  }
  float S = blk_sum256(red, part);
  for (int i = 0; i < 4; ++i) {
    int m = t + i * 256;
    prec_out[b * Mx + m] = (1.0f - S) * pp[b * Mx + m] + wloc[i];
  }
}

// ---------------------------------------------------------------------------
// 8) memory = prev_mem*(1 - ww*erase) + ww*write_vec  (float4 over d)
// ---------------------------------------------------------------------------
__global__ void k_memory(const float* __restrict__ pm, const float* __restrict__ ww,
                         const float* __restrict__ proj, float* __restrict__ mem_out) {
  int gid = blockIdx.x * blockDim.x + threadIdx.x; // B*M*16
  int d4 = gid & 15;
  int bm = gid >> 4;
  int m = bm & (Mx - 1);
  int b = bm >> 10;
  float w = ww[b * Mx + m];
  const float* pr = proj + b * PD;
  float4 pv = ((const float4*)pm)[gid];
  float4 ev = *(const float4*)(pr + 64 + d4 * 4);
  float4 wv = *(const float4*)(pr + d4 * 4);
  float4 o;
  o.x = pv.x * (1.0f - w * ev.x) + w * wv.x;
  o.y = pv.y * (1.0f - w * ev.y) + w * wv.y;
  o.z = pv.z * (1.0f - w * ev.z) + w * wv.z;
  o.w = pv.w * (1.0f - w * ev.w) + w * wv.w;
  ((float4*)mem_out)[gid] = o;
}

__global__ void k_zero(float* __restrict__ p, int n) {
  int gid = blockIdx.x * blockDim.x + threadIdx.x;
  if (gid < n) p[gid] = 0.0f;
}

// ---------------------------------------------------------------------------
// 10) Fused link update + fwd + bwd. 64x64 LDS tile staged via CDNA5 async
//     global->LDS b128 copies (ASYNCcnt) when available, float4 otherwise.
//     All global traffic is b128. fwd[b,r,i]=sum_j pr[r,j]*L[i,j]; bwd via
//     columns. grid = B * 256 tiles, block = 256.
// ---------------------------------------------------------------------------
__global__ void k_link(const float* __restrict__ pl, const float* __restrict__ ww,
                       const float* __restrict__ pp, const float* __restrict__ prw,
                       float* __restrict__ link_out, float* __restrict__ fwd,
                       float* __restrict__ bwd) {
  __shared__ __align__(16) float Lt[64][LSTR];
  __shared__ float wwi[64], wwj[64], pj[64], pri[256], prj[256];
  int blk = blockIdx.x;
  int b = blk >> 8;
  int tile = blk & 255;
  int i0 = (tile >> 4) * 64, j0 = (tile & 15) * 64;
  int t = threadIdx.x;
  if (t < 64) {
    wwi[t] = ww[b * Mx + i0 + t];
    wwj[t] = ww[b * Mx + j0 + t];
    pj[t]  = pp[b * Mx + j0 + t];
  }
  {
    int r = t >> 6, i = t & 63;
    pri[t] = prw[(b * Rx + r) * Mx + i0 + i];
    prj[t] = prw[(b * Rx + r) * Mx + j0 + i];
  }
  const float* plb = pl + (size_t)b * Mx * Mx;
  float* lob = link_out + (size_t)b * Mx * Mx;

#if defined(__gfx1250__) && __has_builtin(__builtin_amdgcn_global_load_async_to_lds_b128)
  // Async DMA: global prev_link tile -> LDS, no VGPR round-trip (ASYNCcnt).
#pragma unroll
  for (int e = 0; e < 4; ++e) {
    int lin4 = t + e * 256;             // 4-element group id
    int i = lin4 >> 4, j4 = (lin4 & 15) * 4;
    const float* gsrc = plb + (size_t)(i0 + i) * Mx + j0 + j4;
    __builtin_amdgcn_global_load_async_to_lds_b128(
        (AS1 v4i_*)gsrc,
        (AS3 v4i_*)&Lt[i][j4], 0, 0);
  }
#if __has_builtin(__builtin_amdgcn_s_wait_asynccnt)
  __builtin_amdgcn_s_wait_asynccnt(0);
#else
  asm volatile("s_wait_asynccnt 0x0" ::: "memory");
#endif
#else
#pragma unroll
  for (int e = 0; e < 4; ++e) {
    int lin4 = t + e * 256;
    int i = lin4 >> 4, j4 = (lin4 & 15) * 4;
    float4 v = *(const float4*)(plb + (size_t)(i0 + i) * Mx + j0 + j4);
    *(float4*)&Lt[i][j4] = v;
  }
#endif
  __syncthreads();

#pragma unroll
  for (int e = 0; e < 4; ++e) {
    int lin4 = t + e * 256;
    int i = lin4 >> 4, j4 = (lin4 & 15) * 4;
    int gi = i0 + i;
    float4 v = *(const float4*)&Lt[i][j4];
    float wi = wwi[i];
    float* vp = &v.x;
#pragma unroll
    for (int q = 0; q < 4; ++q) {
      int gj = j0 + j4 + q;
      float l = (gi == gj) ? 0.0f
                           : (1.0f - wi - wwj[j4 + q]) * vp[q] + wi * pj[j4 + q];
      vp[q] = l;
    }
    *(float4*)&Lt[i][j4] = v;
    *(float4*)(lob + (size_t)gi * Mx + j0 + j4) = v;
  }
  __syncthreads();
  {
    int r = t >> 6, i = t & 63;
    float s = 0.0f;
    for (int j = 0; j < 64; ++j) s += prj[r * 64 + j] * Lt[i][j];
    atomicAdd(&fwd[(b * Rx + r) * Mx + i0 + i], s);
    float s2 = 0.0f;
    for (int i2 = 0; i2 < 64; ++i2) s2 += pri[r * 64 + i2] * Lt[i2][i];
    atomicAdd(&bwd[(b * Rx + r) * Mx + j0 + i], s2);
  }
}

// ---------------------------------------------------------------------------
// 12) read_weights = cm*rcw + fm*fwd + bm*bwd ; read_words = rw @ memory
//     grid = B*R, block = 256
// ---------------------------------------------------------------------------
__global__ void k_read(const float* __restrict__ rcw, const float* __restrict__ fwd,
                       const float* __restrict__ bwd, const float* __restrict__ proj,
                       const float* __restrict__ mem, float* __restrict__ rw_out,
                       float* __restrict__ words_out) {
  __shared__ float rw_s[Mx];
  __shared__ float part[4][Wx];
  int br = blockIdx.x;
  int b = br >> 2, r = br & 3;
  const float* pr = proj + b * PD;
  float bm = pr[134 + r * 3 + 0];
  float fm = pr[134 + r * 3 + 1];
  float cm = pr[134 + r * 3 + 2];
  int t = threadIdx.x;
  for (int m = t; m < Mx; m += 256) {
    float v = cm * rcw[(size_t)br * Mx + m] + fm * fwd[(size_t)br * Mx + m]
            + bm * bwd[(size_t)br * Mx + m];
    rw_s[m] = v;
    rw_out[(size_t)br * Mx + m] = v;
  }
  __syncthreads();
  int g = t >> 6, d = t & 63;
  float acc = 0.0f;
  for (int m = g * 256; m < (g + 1) * 256; ++m)
    acc += rw_s[m] * mem[((size_t)b * Mx + m) * Wx + d];
  part[g][d] = acc;
  __syncthreads();
  if (t < Wx)
    words_out[br * Wx + t] = part[0][t] + part[1][t] + part[2][t] + part[3][t];
}

// ---------------------------------------------------------------------------
extern "C" void kernel_launch(void* const* d_in, const int* in_sizes, int n_in,
                              void* d_out, int out_size, void* d_ws, size_t ws_size,
                              hipStream_t stream) {
  (void)in_sizes; (void)n_in; (void)out_size; (void)ws_size;
  const float* wv_w = (const float*)d_in[0];  const float* wv_b = (const float*)d_in[1];
  const float* ev_w = (const float*)d_in[2];  const float* ev_b = (const float*)d_in[3];
  const float* fg_w = (const float*)d_in[4];  const float* fg_b = (const float*)d_in[5];
  const float* ag_w = (const float*)d_in[6];  const float* ag_b = (const float*)d_in[7];
  const float* wg_w = (const float*)d_in[8];  const float* wg_b = (const float*)d_in[9];
  const float* rm_w = (const float*)d_in[10]; const float* rm_b = (const float*)d_in[11];
  const float* ws_w = (const float*)d_in[12]; const float* ws_b = (const float*)d_in[13];
  const float* rs_w = (const float*)d_in[14]; const float* rs_b = (const float*)d_in[15];
  const float* wk_w = (const float*)d_in[16]; const float* wk_b = (const float*)d_in[17];
  const float* rk_w = (const float*)d_in[18]; const float* rk_b = (const float*)d_in[19];
  const float* x     = (const float*)d_in[20];
  const float* pmem  = (const float*)d_in[21];
  const float* prw   = (const float*)d_in[22];
  const float* pww   = (const float*)d_in[23];
  const float* plink = (const float*)d_in[24];
  const float* pprec = (const float*)d_in[25];
  const float* pusage= (const float*)d_in[26];

  float* out = (float*)d_out;
  float* o_words = out;                       // (B,R,W)      16384
  float* o_mem   = out + 16384;               // (B,M,W)      4194304
  float* o_rw    = out + 4210688;             // (B,R,M)      262144
  float* o_ww    = out + 4472832;             // (B,1,M)      65536
  float* o_link  = out + 4538368;             // (B,1,M,M)    67108864
  float* o_prec  = out + 71647232;            // (B,1,M)      65536
  float* o_usage = out + 71712768;            // (B,M)        65536

  float* W    = (float*)d_ws;
  float* Wc   = W;                            // 512*480 = 245760
  float* bc   = W + 245760;                   // 480
  float* proj = W + 246240;                   // 64*480  = 30720
  float* wcw  = W + 276960;                   // 65536
  float* waw  = W + 342496;                   // 65536
  float* fwd  = W + 408032;                   // 262144
  float* bwd  = W + 670176;                   // 262144
  float* rcw  = W + 932320;                   // 262144

  k_pack<<<962, 256, 0, stream>>>(wv_w, wv_b, ev_w, ev_b, fg_w, fg_b, ag_w, ag_b,
                                  wg_w, wg_b, rm_w, rm_b, ws_w, ws_b, rs_w, rs_b,
                                  wk_w, wk_b, rk_w, rk_b, Wc, bc);
  k_gemm<<<(Bx / 16) * (PD / 16), 32, 0, stream>>>(x, Wc, bc, proj);
  k_act<<<Bx, 128, 0, stream>>>(proj);
  k_usage<<<Bx, 256, 0, stream>>>(pusage, pww, prw, proj, o_usage);
  k_cosine<<<Bx, 256, 0, stream>>>(pmem, proj, wcw, /*heads=*/1, /*key=*/151, /*beta=*/146);
  k_alloc<<<Bx, 1024, 0, stream>>>(o_usage, waw);
  k_ww_prec<<<Bx, 256, 0, stream>>>(proj, waw, wcw, pprec, o_ww, o_prec);
  k_memory<<<(Bx * Mx * Wx / 4) / 256, 256, 0, stream>>>(pmem, o_ww, proj, o_mem);
  k_zero<<<2048, 256, 0, stream>>>(fwd, 2 * Bx * Rx * Mx);   // zeros fwd+bwd (contiguous)
  k_link<<<Bx * 256, 256, 0, stream>>>(plink, o_ww, pprec, prw, o_link, fwd, bwd);
  k_cosine<<<Bx * Rx, 256, 0, stream>>>(o_mem, proj, rcw, /*heads=*/4, /*key=*/215, /*beta=*/147);
  k_read<<<Bx * Rx, 256, 0, stream>>>(rcw, fwd, bwd, proj, o_mem, o_rw, o_words);
}